// MultiModalAttentionLayer_82102594830968
// MI455X (gfx1250) — compile-verified
//
#include <hip/hip_runtime.h>
#include <math.h>

typedef float v2f __attribute__((ext_vector_type(2)));
typedef float v8f __attribute__((ext_vector_type(8)));

#define SEQ   128
#define BATCH 256
#define HID   1024
#define ADIM  1024

#define BM 64           // A-tile rows per workgroup
#define BN 256          // N chunk
#define BKQ 32          // K chunk (Q kernel, single-buffered)
#define BKF 16          // K chunk (fused kernel, double-buffered)
#define A_STRIDE 80     // LDS A chunk transposed [BK][A_STRIDE] (pad 64->80)
#define B_STRIDE 272    // LDS B chunk [BK][B_STRIDE] (pad 256->272)
#define VT_STRIDE 1040  // LDS V tile [64][VT_STRIDE] (pad 1024->1040)

// ---------------------------------------------------------------------------
// gfx1250 async global->LDS copy (ASYNCcnt-tracked, no VGPR round trip).
// Builtin signature (from clang diagnostic): param0 = 16B int-vector pointer
// in AS1 (printed as `__device__`), param1 = LDS-side pointer, then imm
// offset + imm cpol. Falls back to a synchronous b128 copy if unavailable.
// ---------------------------------------------------------------------------
typedef int i32x4 __attribute__((vector_size(4 * sizeof(int))));
typedef __attribute__((address_space(1))) i32x4 gi32x4;
typedef __attribute__((address_space(3))) i32x4 li32x4;

__device__ __forceinline__ void async_copy_b128(const float* gsrc, float* ldst) {
#if defined(__has_builtin) && __has_builtin(__builtin_amdgcn_global_load_async_to_lds_b128)
    __builtin_amdgcn_global_load_async_to_lds_b128(
        (gi32x4*)(void*)gsrc, (li32x4*)ldst, 0, 0);
#else
    *(float4*)ldst = *(const float4*)gsrc;
#endif
}

__device__ __forceinline__ void wait_async0() {
#if defined(__has_builtin) && __has_builtin(__builtin_amdgcn_s_wait_asynccnt)
    __builtin_amdgcn_s_wait_asynccnt(0);
#else
    asm volatile("s_wait_asynccnt 0x0" ::: "memory");
#endif
}

// ---------------------------------------------------------------------------
// Kernel 1: Q = topic @ Wq + bq   ([256,1024] = [256,1024] @ [1024,1024])
// grid = (ADIM/BN, BATCH/BM) = (4,4), block = 256 (8 waves)
// ---------------------------------------------------------------------------
__global__ __launch_bounds__(256) void q_gemm_kernel(
    const float* __restrict__ topic, const float* __restrict__ Wq,
    const float* __restrict__ bq, float* __restrict__ Q)
{
    extern __shared__ float lds[];
    float* As = lds;                     // [BKQ][A_STRIDE] transposed (k-major)
    float* Bs = As + BKQ * A_STRIDE;     // [BKQ][B_STRIDE]

    const int nc = blockIdx.x * BN;
    const int b0 = blockIdx.y * BM;
    const int t  = threadIdx.x;
    const int lane = t & 31, wave = t >> 5;
    const int l16 = lane & 15, lh = lane >> 4;
    const int mt = (wave & 3) * 16;
    const int nb = (wave >> 2) * 128;

    v8f acc[8] = {};

    #pragma unroll 1
    for (int kc = 0; kc < HID; kc += BKQ) {
        __syncthreads();                         // prev chunk fully consumed
        #pragma unroll
        for (int i = 0; i < 8; ++i) {            // A chunk (transposed, sync)
            int idx = t + i * 256;
            int m = idx >> 5;
            int k = idx & 31;
            As[k * A_STRIDE + m] = topic[(size_t)(b0 + m) * HID + kc + k];
        }
        #pragma unroll
        for (int i = 0; i < 8; ++i) {            // B chunk (async -> LDS)
            int kr = (t >> 6) + i * 4;
            int c4 = (t & 63) * 4;
            async_copy_b128(Wq + (size_t)(kc + kr) * ADIM + nc + c4,
                            Bs + kr * B_STRIDE + c4);
        }
        wait_async0();
        __syncthreads();
        #pragma unroll
        for (int k = 0; k < BKQ; k += 4) {
            const int ak = k + lh * 2;
            v2f a;
            a.x = As[ak * A_STRIDE + mt + l16];
            a.y = As[(ak + 1) * A_STRIDE + mt + l16];
            #pragma unroll
            for (int j = 0; j < 8; ++j) {
                const int n = nb + j * 16 + l16;
                v2f b;
                b.x = Bs[ak * B_STRIDE + n];
                b.y = Bs[(ak + 1) * B_STRIDE + n];
                acc[j] = __builtin_amdgcn_wmma_f32_16x16x4_f32(
                    false, a, false, b, (short)0, acc[j], false, false);
            }
        }
    }
    #pragma unroll
    for (int j = 0; j < 8; ++j) {
        const int col = nc + nb + j * 16 + l16;
        const float bias = bq[col];
        #pragma unroll
        for (int r = 0; r < 8; ++r) {
            const int row = b0 + mt + lh * 8 + r;
            Q[(size_t)row * ADIM + col] = acc[j][r] + bias;
        }
    }
}

// ---------------------------------------------------------------------------
// Kernel 2: fused per (s, 32-batch tile); double-buffered async staging.
//   V0 = topic_tile @ Wv[s] + bv[s]   (rows  0..31 of LDS V tile)
//   V1 = text[s]_tile @ Wv[s] + bv[s] (rows 32..63 of LDS V tile)
//   s0/s1 = Q . V0/V1 / sqrt(d); 2-way softmax; out = a0*V0 + a1*V1
// grid = (BATCH/32, SEQ) = (8,128), block = 256 (8 waves)
// ---------------------------------------------------------------------------
__global__ __launch_bounds__(256) void fused_attn_kernel(
    const float* __restrict__ topic,  // [256,1024]
    const float* __restrict__ text,   // [128,256,1024]
    const float* __restrict__ Wv,     // [128,1024,1024]
    const float* __restrict__ bv,     // [128,1024]
    const float* __restrict__ Q,      // [256,1024] (workspace)
    float* __restrict__ out)          // [128,256,1024]
{
    extern __shared__ float lds[];
    float* Vt  = lds;                           // [64][VT_STRIDE]
    float* As  = Vt + 64 * VT_STRIDE;           // 2 x [BKF][A_STRIDE]
    float* Bs  = As + 2 * BKF * A_STRIDE;       // 2 x [BKF][B_STRIDE]
    float* scr = Bs + 2 * BKF * B_STRIDE;       // [64][4] logit partials
    float* wgt = scr + 64 * 4;                  // [64] softmax weights

    const int s  = blockIdx.y;
    const int b0 = blockIdx.x * 32;
    const int t  = threadIdx.x;
    const int lane = t & 31, wave = t >> 5;
    const int l16 = lane & 15, lh = lane >> 4;
    const int mt = (wave & 3) * 16;
    const int nb = (wave >> 2) * 128;

    const float* Wvs = Wv + (size_t)s * HID * ADIM;
    const float* bvs = bv + (size_t)s * ADIM;

    // stage one K-chunk (A sync-transposed, B async) into buffer `buf`
    auto stage = [&](int kc, int nc, int buf) {
        float* Asb = As + buf * (BKF * A_STRIDE);
        float* Bsb = Bs + buf * (BKF * B_STRIDE);
        #pragma unroll
        for (int i = 0; i < 4; ++i) {            // A: 64 rows x 16 k
            int idx = t + i * 256;
            int m = idx >> 4;                    // rows; no wave straddles m=32
            int k = idx & 15;
            const float* src = (m < 32)
                ? topic + (size_t)(b0 + m) * HID + kc + k
                : text + ((size_t)s * BATCH + b0 + m - 32) * HID + kc + k;
            Asb[k * A_STRIDE + m] = *src;
        }
        #pragma unroll
        for (int i = 0; i < 4; ++i) {            // B: 16 rows x 256 cols
            int kr = (t >> 6) + i * 4;
            int c4 = (t & 63) * 4;
            async_copy_b128(Wvs + (size_t)(kc + kr) * ADIM + nc + c4,
                            Bsb + kr * B_STRIDE + c4);
        }
    };

    // ---- Phase 1: build 64x1024 V tile in LDS via fp32 WMMA --------------
    #pragma unroll 1
    for (int nc = 0; nc < ADIM; nc += BN) {
        v8f acc[8] = {};

        stage(0, nc, 0);                         // prologue
        wait_async0();
        __syncthreads();

        #pragma unroll 1
        for (int kc = 0; kc < HID; kc += BKF) {
            const int cur = (kc / BKF) & 1;
            if (kc + BKF < HID)                  // prefetch next chunk (async)
                stage(kc + BKF, nc, cur ^ 1);

            const float* Asb = As + cur * (BKF * A_STRIDE);
            const float* Bsb = Bs + cur * (BKF * B_STRIDE);
            #pragma unroll
            for (int k = 0; k < BKF; k += 4) {
                const int ak = k + lh * 2;
                v2f a;
                a.x = Asb[ak * A_STRIDE + mt + l16];
                a.y = Asb[(ak + 1) * A_STRIDE + mt + l16];
                #pragma unroll
                for (int j = 0; j < 8; ++j) {
                    const int n = nb + j * 16 + l16;
                    v2f b;
                    b.x = Bsb[ak * B_STRIDE + n];
                    b.y = Bsb[(ak + 1) * B_STRIDE + n];
                    acc[j] = __builtin_amdgcn_wmma_f32_16x16x4_f32(
                        false, a, false, b, (short)0, acc[j], false, false);
                }
            }
            wait_async0();                       // next-chunk copies landed
            __syncthreads();                     // everyone done with `cur`
        }

        #pragma unroll
        for (int j = 0; j < 8; ++j) {            // + bias, spill into V tile
            const int ncol = nc + nb + j * 16 + l16;
            const float bias = bvs[ncol];
            #pragma unroll
            for (int r = 0; r < 8; ++r) {
                const int row = mt + lh * 8 + r;
                Vt[row * VT_STRIDE + ncol] = acc[j][r] + bias;
            }
        }
    }
    __syncthreads();

    // ---- Phase 2: logits s0/s1 = Q . V, 4 threads per row ----------------
    {
        const int r = t >> 2;                 // 0..63 (V-tile row)
        const int p = t & 3;                  // quarter of d
        const int b = r & 31;                 // Q row depends only on batch
        const float* qrow = Q + (size_t)(b0 + b) * ADIM + p * 256;
        const float* vrow = Vt + r * VT_STRIDE + p * 256;
        float sum = 0.f;
        #pragma unroll 4
        for (int i = 0; i < 256; i += 4) {
            float4 q4 = *(const float4*)(qrow + i);
            float4 v4 = *(const float4*)(vrow + i);
            sum += q4.x * v4.x + q4.y * v4.y + q4.z * v4.z + q4.w * v4.w;
        }
        scr[r * 4 + p] = sum;
    }
    __syncthreads();

    // ---- stable 2-way softmax per batch row ------------------------------
    if (t < 32) {
        const float scale = 0.03125f;         // 1/sqrt(1024)
        float s0 = (scr[t * 4] + scr[t * 4 + 1] + scr[t * 4 + 2] + scr[t * 4 + 3]) * scale;
        int r1 = 32 + t;
        float s1 = (scr[r1 * 4] + scr[r1 * 4 + 1] + scr[r1 * 4 + 2] + scr[r1 * 4 + 3]) * scale;
        float m  = fmaxf(s0, s1);
        float e0 = __expf(s0 - m);
        float e1 = __expf(s1 - m);
        float inv = 1.f / (e0 + e1);
        wgt[t]      = e0 * inv;
        wgt[32 + t] = e1 * inv;
    }
    __syncthreads();

    // ---- Phase 3: out = a0*V0 + a1*V1 ------------------------------------
    for (int idx = t * 4; idx < 32 * ADIM; idx += 256 * 4) {
        const int row = idx >> 10;
        const int d   = idx & 1023;
        const float a0 = wgt[row];
        const float a1 = wgt[32 + row];
        float4 v0 = *(const float4*)(Vt + row * VT_STRIDE + d);
        float4 v1 = *(const float4*)(Vt + (32 + row) * VT_STRIDE + d);
        float4 o;
        o.x = a0 * v0.x + a1 * v1.x;
        o.y = a0 * v0.y + a1 * v1.y;
        o.z = a0 * v0.z + a1 * v1.z;
        o.w = a0 * v0.w + a1 * v1.w;
        *(float4*)(out + ((size_t)s * BATCH + b0 + row) * ADIM + d) = o;
    }
}

// ---------------------------------------------------------------------------
extern "C" void kernel_launch(void* const* d_in, const int* in_sizes, int n_in,
                              void* d_out, int out_size, void* d_ws, size_t ws_size,
                              hipStream_t stream) {
    const float* topic = (const float*)d_in[0];  // [1,256,1024]
    const float* text  = (const float*)d_in[1];  // [128,256,1024]
    const float* Wq    = (const float*)d_in[2];  // [1024,1024]
    const float* bq    = (const float*)d_in[3];  // [1024]
    const float* Wv    = (const float*)d_in[4];  // [128,1024,1024]
    const float* bv    = (const float*)d_in[5];  // [128,1024]
    float* out = (float*)d_out;                  // [128,256,1024]
    float* Q   = (float*)d_ws;                   // [256,1024] scratch

    const size_t lds1 = (size_t)(BKQ * A_STRIDE + BKQ * B_STRIDE) * sizeof(float);
    const size_t lds2 = (size_t)(64 * VT_STRIDE + 2 * BKF * A_STRIDE
                                 + 2 * BKF * B_STRIDE + 64 * 4 + 64) * sizeof(float);

    (void)hipFuncSetAttribute(reinterpret_cast<const void*>(&q_gemm_kernel),
                              hipFuncAttributeMaxDynamicSharedMemorySize, (int)lds1);
    (void)hipFuncSetAttribute(reinterpret_cast<const void*>(&fused_attn_kernel),
                              hipFuncAttributeMaxDynamicSharedMemorySize, (int)lds2);

    q_gemm_kernel<<<dim3(ADIM / BN, BATCH / BM), 256, lds1, stream>>>(topic, Wq, bq, Q);
    fused_attn_kernel<<<dim3(BATCH / 32, SEQ), 256, lds2, stream>>>(topic, text, Wv, bv, Q, out);
}